// GCN_15908558865647
// MI455X (gfx1250) — compile-verified
//
#include <hip/hip_runtime.h>
#include <hip/hip_bf16.h>

typedef __attribute__((ext_vector_type(2))) float v2f;
typedef __attribute__((ext_vector_type(8))) float v8f;

#define N_NODES 40000
#define N_EDGES 640000
#define D 128

// ---------------------------------------------------------------------------
// deg[i] = 1.0 (self loop weight)
__global__ void gcn_init_deg(float* __restrict__ deg) {
    int i = blockIdx.x * blockDim.x + threadIdx.x;
    if (i < N_NODES) deg[i] = 1.0f;
}

// deg[dst[e]] += ew[e]
__global__ void gcn_deg_scatter(const long long* __restrict__ ei,
                                const float* __restrict__ ew,
                                float* __restrict__ deg) {
    int e = blockIdx.x * blockDim.x + threadIdx.x;
    if (e < N_EDGES) {
        int d = (int)ei[N_EDGES + e];
        atomicAdd(&deg[d], ew[e]);
    }
}

// dinv = rsqrt(deg) (deg >= 1 always), selfc = dinv*dinv
__global__ void gcn_dinv(float* __restrict__ dinv_inplace_deg,
                         float* __restrict__ selfc) {
    int i = blockIdx.x * blockDim.x + threadIdx.x;
    if (i < N_NODES) {
        float dg = dinv_inplace_deg[i];
        float r = (dg > 0.0f) ? rsqrtf(dg) : 0.0f;
        dinv_inplace_deg[i] = r;
        selfc[i] = r * r;
    }
}

// norm[e] = dinv[src]*ew*dinv[dst]
__global__ void gcn_edge_norm(const long long* __restrict__ ei,
                              const float* __restrict__ ew,
                              const float* __restrict__ dinv,
                              float* __restrict__ norm) {
    int e = blockIdx.x * blockDim.x + threadIdx.x;
    if (e < N_EDGES) {
        int s = (int)ei[e];
        int d = (int)ei[N_EDGES + e];
        norm[e] = dinv[s] * ew[e] * dinv[d];
    }
}

// ---------------------------------------------------------------------------
// H = Z @ W   (Z: [N_NODES x 128] row-major, W: [128 x 128] row-major)
// grid.x = N_NODES/16 blocks; 256 threads = 8 wave32; wave w owns cols
// [16w, 16w+16), block owns rows [16*bx, 16*bx+16).
// Uses exact-FP32 V_WMMA_F32_16X16X4_F32, K chained 32 times.
__global__ void __launch_bounds__(256)
gcn_gemm_wmma(const float* __restrict__ Z, const float* __restrict__ W,
              float* __restrict__ H) {
    const int wave = threadIdx.x >> 5;
    const int lane = threadIdx.x & 31;
    const int r  = lane & 15;   // row of A / col of B / N index of C
    const int hi = lane >> 4;   // K-half selector (A/B), M-half selector (C/D)

    const int m0 = blockIdx.x * 16;
    const int n0 = wave * 16;

    const float* __restrict__ zrow = Z + (size_t)(m0 + r) * D;

    v8f c = {0.f, 0.f, 0.f, 0.f, 0.f, 0.f, 0.f, 0.f};

#pragma unroll
    for (int kb = 0; kb < D; kb += 4) {
        const int k = kb + 2 * hi;
        // A 16x4: lane r holds A[m0+r][k], A[m0+r][k+1]
        v2f a = *(const v2f*)(zrow + k);
        // B 4x16: lane r holds B[k][n0+r], B[k+1][n0+r]
        v2f b;
        b.x = W[(size_t)k * D + n0 + r];
        b.y = W[(size_t)(k + 1) * D + n0 + r];
        c = __builtin_amdgcn_wmma_f32_16x16x4_f32(
            /*neg_a=*/false, a, /*neg_b=*/false, b,
            /*c_mod=*/(short)0, c, /*reuse_a=*/false, /*reuse_b=*/false);
    }

    // D layout: VGPR v -> C[v + 8*hi][r]
    float* __restrict__ hout = H + n0 + r;
#pragma unroll
    for (int v = 0; v < 8; ++v) {
        hout[(size_t)(m0 + 8 * hi + v) * D] = c[v];
    }
}

// ---------------------------------------------------------------------------
// agg[i][:] = selfc[i] * h[i][:]   (self-loop term; fully initializes agg)
__global__ void gcn_agg_self(const float* __restrict__ selfc,
                             const float* __restrict__ H,
                             float* __restrict__ AGG) {
    int gid = blockIdx.x * blockDim.x + threadIdx.x;  // N_NODES*32 threads
    int i = gid >> 5;
    int f = (gid & 31) * 4;
    if (i < N_NODES) {
        float s = selfc[i];
        const float4 h4 = *(const float4*)(H + (size_t)i * D + f);
        float4 o;
        o.x = s * h4.x; o.y = s * h4.y; o.z = s * h4.z; o.w = s * h4.w;
        *(float4*)(AGG + (size_t)i * D + f) = o;
    }
}

// agg[dst] += norm[e] * h[src]  (32 threads per edge, 4 floats each)
__global__ void gcn_scatter(const long long* __restrict__ ei,
                            const float* __restrict__ norm,
                            const float* __restrict__ H,
                            float* __restrict__ AGG) {
    int gid = blockIdx.x * blockDim.x + threadIdx.x;  // N_EDGES*32 threads
    int e = gid >> 5;
    int f = (gid & 31) * 4;
    if (e < N_EDGES) {
        int s = (int)ei[e];
        int d = (int)ei[N_EDGES + e];
        float w = norm[e];
        const float4 h4 = *(const float4*)(H + (size_t)s * D + f);
        float* out = AGG + (size_t)d * D + f;
        atomicAdd(out + 0, w * h4.x);
        atomicAdd(out + 1, w * h4.y);
        atomicAdd(out + 2, w * h4.z);
        atomicAdd(out + 3, w * h4.w);
    }
}

// z = relu(agg + b)
__global__ void gcn_bias_relu(const float* __restrict__ AGG,
                              const float* __restrict__ bias,
                              float* __restrict__ ZOUT) {
    int gid = blockIdx.x * blockDim.x + threadIdx.x;  // N_NODES*32 threads
    int i = gid >> 5;
    int f = (gid & 31) * 4;
    if (i < N_NODES) {
        const float4 a4 = *(const float4*)(AGG + (size_t)i * D + f);
        const float4 b4 = *(const float4*)(bias + f);
        float4 o;
        o.x = fmaxf(a4.x + b4.x, 0.0f);
        o.y = fmaxf(a4.y + b4.y, 0.0f);
        o.z = fmaxf(a4.z + b4.z, 0.0f);
        o.w = fmaxf(a4.w + b4.w, 0.0f);
        *(float4*)(ZOUT + (size_t)i * D + f) = o;
    }
}

// ---------------------------------------------------------------------------
extern "C" void kernel_launch(void* const* d_in, const int* in_sizes, int n_in,
                              void* d_out, int out_size, void* d_ws, size_t ws_size,
                              hipStream_t stream) {
    const float*      x  = (const float*)d_in[0];
    const long long*  ei = (const long long*)d_in[1];   // int64 [2, E]
    const float*      ew = (const float*)d_in[2];
    const float*      W1 = (const float*)d_in[3];
    const float*      b1 = (const float*)d_in[4];
    const float*      W2 = (const float*)d_in[5];
    const float*      b2 = (const float*)d_in[6];
    const float*      W3 = (const float*)d_in[7];
    const float*      b3 = (const float*)d_in[8];
    float* out = (float*)d_out;

    float* ws    = (float*)d_ws;
    float* dinv  = ws;                       // 40000 (deg -> dinv in place)
    float* selfc = dinv  + N_NODES;          // 40000
    float* norm  = selfc + N_NODES;          // 640000
    float* H     = norm  + N_EDGES;          // 5,120,000
    float* AGG   = H     + (size_t)N_NODES * D;  // 5,120,000
    float* ZA    = AGG   + (size_t)N_NODES * D;  // 5,120,000

    const int node_blk  = (N_NODES + 255) / 256;              // 157
    const int edge_blk  = (N_EDGES + 255) / 256;              // 2500
    const int nfeat_blk = ((size_t)N_NODES * 32 + 255) / 256; // 5000
    const int efeat_blk = ((size_t)N_EDGES * 32 + 255) / 256; // 80000
    const int gemm_blk  = N_NODES / 16;                       // 2500

    // normalization
    gcn_init_deg<<<node_blk, 256, 0, stream>>>(dinv);
    gcn_deg_scatter<<<edge_blk, 256, 0, stream>>>(ei, ew, dinv);
    gcn_dinv<<<node_blk, 256, 0, stream>>>(dinv, selfc);
    gcn_edge_norm<<<edge_blk, 256, 0, stream>>>(ei, ew, dinv, norm);

    // layer 1
    gcn_gemm_wmma<<<gemm_blk, 256, 0, stream>>>(x, W1, H);
    gcn_agg_self<<<nfeat_blk, 256, 0, stream>>>(selfc, H, AGG);
    gcn_scatter<<<efeat_blk, 256, 0, stream>>>(ei, norm, H, AGG);
    gcn_bias_relu<<<nfeat_blk, 256, 0, stream>>>(AGG, b1, ZA);

    // layer 2
    gcn_gemm_wmma<<<gemm_blk, 256, 0, stream>>>(ZA, W2, H);
    gcn_agg_self<<<nfeat_blk, 256, 0, stream>>>(selfc, H, AGG);
    gcn_scatter<<<efeat_blk, 256, 0, stream>>>(ei, norm, H, AGG);
    gcn_bias_relu<<<nfeat_blk, 256, 0, stream>>>(AGG, b2, ZA);

    // layer 3
    gcn_gemm_wmma<<<gemm_blk, 256, 0, stream>>>(ZA, W3, H);
    gcn_agg_self<<<nfeat_blk, 256, 0, stream>>>(selfc, H, AGG);
    gcn_scatter<<<efeat_blk, 256, 0, stream>>>(ei, norm, H, AGG);
    gcn_bias_relu<<<nfeat_blk, 256, 0, stream>>>(AGG, b3, out);
}